// YOLO_V3_Loss_46205258170791
// MI455X (gfx1250) — compile-verified
//
#include <hip/hip_runtime.h>
#include <hip/hip_bf16.h>

typedef float v2f __attribute__((ext_vector_type(2)));
typedef float v4f __attribute__((ext_vector_type(4)));
typedef float v8f __attribute__((ext_vector_type(8)));

#define MAXBLK 1536

// Full-wave (32 lane) f32 sum on the CDNA5 matrix pipe:
// two V_WMMA_F32_16X16X4_F32 with a ones B-matrix.
// A layout (16x4 f32, 2 VGPRs): a[0] lane L<16 -> (M=L,K=0), lane L>=16 -> (M=L-16,K=2).
// With a[1]=0, D[m,n] = v[m] + v[m+16]. Summing the 8 D VGPRs gives the sum of
// rows 0..7 on lanes 0-15 and rows 8..15 on lanes 16-31; a second WMMA folds
// the two half-sums so every lane of d2[0] holds the full 32-lane total.
// EXEC must be all ones at every call site.
__device__ __forceinline__ float wave_sum32(float v) {
    v2f a; a[0] = v;    a[1] = 0.0f;
    v2f b; b[0] = 1.0f; b[1] = 1.0f;
    v8f c = {};
    v8f d = __builtin_amdgcn_wmma_f32_16x16x4_f32(false, a, false, b,
                                                  (short)0, c, false, false);
    float t = (d[0] + d[1]) + (d[2] + d[3]) + ((d[4] + d[5]) + (d[6] + d[7]));
    v2f a2; a2[0] = t; a2[1] = 0.0f;
    v8f d2 = __builtin_amdgcn_wmma_f32_16x16x4_f32(false, a2, false, b,
                                                   (short)0, c, false, false);
    return d2[0];
}

// Per-anchor positive count + last positive index (flat anchor index).
__global__ void count_pos_kernel(const unsigned char* __restrict__ mpos,
                                 unsigned int* __restrict__ c, int n) {
    int i = blockIdx.x * blockDim.x + threadIdx.x;
    if (i < n && mpos[i]) {
        atomicAdd(&c[0], 1u);            // n_pos
        atomicMax(&c[2], (unsigned)i);   // idx_last
    }
}

// One streaming pass over pred/gt/mask_neg for one scale.
// partials[block*4 + {0,1,2,3}] = {coord, conf, bce, neg} block sums.
__global__ void __launch_bounds__(256) scale_main_kernel(
    const float* __restrict__ pred, const float* __restrict__ gt,
    const unsigned char* __restrict__ mpos, const unsigned char* __restrict__ mneg,
    float* __restrict__ partials, unsigned int* __restrict__ nneg_out, int n) {

    float coord = 0.f, conf = 0.f, bce = 0.f, neg = 0.f;
    unsigned int negcnt = 0u;

    const int tid    = blockIdx.x * blockDim.x + threadIdx.x;
    const int stride = gridDim.x * blockDim.x;
    const int n4     = n >> 2;

    const v4f* __restrict__ p4 = (const v4f*)pred;
    const v4f* __restrict__ g4 = (const v4f*)gt;
    const unsigned int* __restrict__ mn4p = (const unsigned int*)mneg;

    for (int i = tid; i < n4; i += stride) {
        v4f p = p4[i];
        v4f g = g4[i];
        unsigned int mn4 = mn4p[i];
        int e0 = i << 2;
#pragma unroll
        for (int k = 0; k < 4; ++k) {
            unsigned int e  = (unsigned int)(e0 + k);
            unsigned int a  = e / 85u;
            unsigned int ch = e - a * 85u;
            float pv = p[k], gv = g[k];
            float dlt = gv - pv;
            float d2  = dlt * dlt;
            if ((mn4 >> (8 * k)) & 0xFFu) { neg += d2; ++negcnt; }
            if (mpos[a]) {
                if (ch < 4u)       coord += d2;
                else if (ch == 4u) conf  += d2;
                else bce += fmaxf(pv, 0.f) - pv * gv + log1pf(expf(-fabsf(pv)));
            }
        }
    }
    // scalar tail (n % 4) — zero iterations for these shapes, kept for safety
    for (int e = (n4 << 2) + tid; e < n; e += stride) {
        unsigned int a  = (unsigned)e / 85u;
        unsigned int ch = (unsigned)e - a * 85u;
        float pv = pred[e], gv = gt[e];
        float dlt = gv - pv, d2 = dlt * dlt;
        if (mneg[e]) { neg += d2; ++negcnt; }
        if (mpos[a]) {
            if (ch < 4u)       coord += d2;
            else if (ch == 4u) conf  += d2;
            else bce += fmaxf(pv, 0.f) - pv * gv + log1pf(expf(-fabsf(pv)));
        }
    }

    // Deterministic block reduction: WMMA wave reduce, then 8 waves via LDS.
    __shared__ float wsum[8][4];
    __shared__ unsigned int wneg;
    if (threadIdx.x == 0) wneg = 0u;
    __syncthreads();

    float r0 = wave_sum32(coord);
    float r1 = wave_sum32(conf);
    float r2 = wave_sum32(bce);
    float r3 = wave_sum32(neg);

    const int wave = threadIdx.x >> 5, lane = threadIdx.x & 31;
    if (lane == 0) {
        wsum[wave][0] = r0; wsum[wave][1] = r1;
        wsum[wave][2] = r2; wsum[wave][3] = r3;
    }
    if (negcnt) atomicAdd(&wneg, negcnt);
    __syncthreads();

    if (threadIdx.x == 0) {
        float s0 = 0, s1 = 0, s2 = 0, s3 = 0;
        for (int w = 0; w < 8; ++w) {
            s0 += wsum[w][0]; s1 += wsum[w][1];
            s2 += wsum[w][2]; s3 += wsum[w][3];
        }
        float* o = partials + (size_t)blockIdx.x * 4;
        o[0] = s0; o[1] = s1; o[2] = s2; o[3] = s3;
        atomicAdd(nneg_out, wneg);   // integer: deterministic
    }
}

__global__ void __launch_bounds__(256) finalize_kernel(
    const float* __restrict__ partials, const unsigned int* __restrict__ cnts,
    const float* __restrict__ mod0, const float* __restrict__ mod1,
    const float* __restrict__ mod2,
    int nb0, int nb1, int nb2, float* __restrict__ out) {

    __shared__ float lds[8][12];
    float acc[12];
#pragma unroll
    for (int j = 0; j < 12; ++j) acc[j] = 0.f;

    int nb[3] = {nb0, nb1, nb2};
    for (int s = 0; s < 3; ++s) {
        const float* base = partials + (size_t)s * MAXBLK * 4;
        for (int b = threadIdx.x; b < nb[s]; b += 256) {
#pragma unroll
            for (int j = 0; j < 4; ++j) acc[s * 4 + j] += base[(size_t)b * 4 + j];
        }
    }

    const int wave = threadIdx.x >> 5, lane = threadIdx.x & 31;
#pragma unroll
    for (int j = 0; j < 12; ++j) {
        float t = wave_sum32(acc[j]);   // EXEC full here
        if (lane == 0) lds[wave][j] = t;
    }
    __syncthreads();

    if (threadIdx.x == 0) {
        const float* mods[3] = {mod0, mod1, mod2};
        float total = 0.f;
        for (int s = 0; s < 3; ++s) {
            float S0 = 0, S1 = 0, S2 = 0, S3 = 0;
            for (int w = 0; w < 8; ++w) {
                S0 += lds[w][s * 4 + 0]; S1 += lds[w][s * 4 + 1];
                S2 += lds[w][s * 4 + 2]; S3 += lds[w][s * 4 + 3];
            }
            unsigned int npos = cnts[s * 3 + 0];
            unsigned int nneg = cnts[s * 3 + 1];
            unsigned int il   = cnts[s * 3 + 2];
            float wl  = mods[s][(size_t)il * 6];         // modulus[idx_last, 0]
            float npf = (float)(npos > 0u ? npos : 1u);
            float pos_loss = (npos > 0u)
                ? (wl * S0 + S1) / npf + S2 / (npf * 80.0f)   // C = 80
                : 0.f;
            float nnf = (float)(nneg > 0u ? nneg : 1u);
            float neg_loss = (nneg > 0u) ? 3.0f * S3 / nnf : 0.f;  // L_NOOBJ = 3
            total += pos_loss + neg_loss;
        }
        out[0] = total;
    }
}

extern "C" void kernel_launch(void* const* d_in, const int* in_sizes, int n_in,
                              void* d_out, int out_size, void* d_ws, size_t ws_size,
                              hipStream_t stream) {
    // input order: {small,middle,big} x {pred, gt, modulus, mask_pos, mask_neg}
    const float* pred[3] = {(const float*)d_in[0], (const float*)d_in[5],  (const float*)d_in[10]};
    const float* gtp [3] = {(const float*)d_in[1], (const float*)d_in[6],  (const float*)d_in[11]};
    const float* mod [3] = {(const float*)d_in[2], (const float*)d_in[7],  (const float*)d_in[12]};
    const unsigned char* mpos[3] = {(const unsigned char*)d_in[3],
                                    (const unsigned char*)d_in[8],
                                    (const unsigned char*)d_in[13]};
    const unsigned char* mneg[3] = {(const unsigned char*)d_in[4],
                                    (const unsigned char*)d_in[9],
                                    (const unsigned char*)d_in[14]};
    int nelem[3] = {in_sizes[0], in_sizes[5], in_sizes[10]};
    int nanch[3] = {in_sizes[3], in_sizes[8], in_sizes[13]};

    float* partials = (float*)d_ws;
    const size_t CNT_OFF = (size_t)3 * MAXBLK * 4 * sizeof(float);
    unsigned int* cnts = (unsigned int*)((char*)d_ws + CNT_OFF);
    // counters are accumulated with atomics -> zero them every call
    hipMemsetAsync(cnts, 0, 9 * sizeof(unsigned int), stream);

    int nb[3];
    for (int s = 0; s < 3; ++s) {
        int n4 = nelem[s] >> 2;
        int b = (n4 + 2047) / 2048;      // ~8 float4 iterations per thread
        if (b > MAXBLK) b = MAXBLK;
        if (b < 1) b = 1;
        nb[s] = b;
        count_pos_kernel<<<(nanch[s] + 255) / 256, 256, 0, stream>>>(
            mpos[s], cnts + s * 3, nanch[s]);
        scale_main_kernel<<<b, 256, 0, stream>>>(
            pred[s], gtp[s], mpos[s], mneg[s],
            partials + (size_t)s * MAXBLK * 4, cnts + s * 3 + 1, nelem[s]);
    }
    finalize_kernel<<<1, 256, 0, stream>>>(partials, cnts,
                                           mod[0], mod[1], mod[2],
                                           nb[0], nb[1], nb[2], (float*)d_out);
}